// MergedColumnParallelLinearWithTopping_63926293234283
// MI455X (gfx1250) — compile-verified
//
#include <hip/hip_runtime.h>
#include <hip/hip_bf16.h>

// Problem constants (from the reference)
constexpr int S    = 4096;   // tokens
constexpr int H    = 4096;   // hidden
constexpr int NAD  = 4;      // adapters
constexpr int RANK = 16;
constexpr int BDIM = 4096;
constexpr int OUT  = 2 * BDIM;   // 8192

// Tile config
constexpr int TM = 128;   // rows per workgroup tile
constexpr int TN = 128;   // cols per workgroup tile
constexpr int KB = 32;    // K-slab per LDS stage

typedef float v8f __attribute__((ext_vector_type(8)));
typedef float v2f __attribute__((ext_vector_type(2)));

__device__ __forceinline__ v8f wmma4(v2f a, v2f b, v8f c) {
  // D = A(16x4, f32) * B(4x16, f32) + C(16x16, f32)
  return __builtin_amdgcn_wmma_f32_16x16x4_f32(
      /*neg_a=*/false, a, /*neg_b=*/false, b,
      /*c_mod=*/(short)0, c, /*reuse_a=*/false, /*reuse_b=*/false);
}

// ---------------------------------------------------------------- utilities
__global__ void k_zero(int* counts, int* cursor) {
  int t = threadIdx.x;
  if (t < NAD) { counts[t] = 0; cursor[t] = 0; }
}

__global__ void k_count(const int* __restrict__ widx, int* counts) {
  int s = blockIdx.x * blockDim.x + threadIdx.x;
  if (s < S) atomicAdd(&counts[widx[s]], 1);
}

__global__ void k_scan(const int* __restrict__ counts, int* offs) {
  if (threadIdx.x == 0) {
    int acc = 0;
    for (int n = 0; n < NAD; ++n) { offs[n] = acc; acc += counts[n]; }
    offs[NAD] = acc;
  }
}

__global__ void k_scatter(const int* __restrict__ widx, const int* __restrict__ offs,
                          int* cursor, int* perm) {
  int s = blockIdx.x * blockDim.x + threadIdx.x;
  if (s < S) {
    int n = widx[s];
    int pos = offs[n] + atomicAdd(&cursor[n], 1);
    perm[pos] = s;
  }
}

// ------------------------------------------------ per-token LoRA-A + rowsum
// xa[s, 0..31] = x[s,:] @ A[n,:,:]   (both halves), xsum[s] = sum_h x[s,h]
__global__ __launch_bounds__(256) void k_prep(const float* __restrict__ x,
                                              const int* __restrict__ widx,
                                              const float* __restrict__ Abuf,
                                              float* __restrict__ xa,
                                              float* __restrict__ xsum) {
  int s = blockIdx.x;
  int tid = threadIdx.x;
  int n = widx[s];
  float part[33];
#pragma unroll
  for (int r = 0; r < 33; ++r) part[r] = 0.f;
  const float* xrow = x + (size_t)s * H;
  const float* Abase = Abuf + (size_t)n * H * (2 * RANK);
  for (int h = tid; h < H; h += 256) {
    float xv = xrow[h];
    part[32] += xv;
    const float* ap = Abase + (size_t)h * (2 * RANK);
#pragma unroll
    for (int r = 0; r < 32; ++r) part[r] += xv * ap[r];
  }
  __shared__ float sm[33 * 256];
#pragma unroll
  for (int r = 0; r < 33; ++r) sm[r * 256 + tid] = part[r];
  __syncthreads();
  if (tid < 33) {
    float acc = 0.f;
    const float* p = &sm[tid * 256];
    for (int t = 0; t < 256; ++t) acc += p[t];
    if (tid < 32) xa[(size_t)s * 32 + tid] = acc;
    else          xsum[s] = acc;
  }
}

// --------------------------------------------------------- main fused GEMM
// grid = (OUT/TN, S/TM, NAD); block = 256 (8 wave32)
// LDS tiles are pair-interleaved along K: element (k, i) lives at
// [k>>1][i][k&1], so each WMMA operand (k, k+1) is one aligned ds_load_b64.
__global__ __launch_bounds__(256) void k_main(
    const float* __restrict__ x, const int* __restrict__ perm,
    const int* __restrict__ offs, const float* __restrict__ bw,
    const float* __restrict__ Bbuf, const float* __restrict__ DW,
    const float* __restrict__ metas, const float* __restrict__ ssp,
    const float* __restrict__ xa, const float* __restrict__ xsum,
    float* __restrict__ out) {
  const int n  = blockIdx.z;
  const int g0 = offs[n], g1 = offs[n + 1];
  const int r0 = g0 + blockIdx.y * TM;
  if (r0 >= g1) return;                       // tile past this adapter group
  const int jbase = blockIdx.x * TN;

  const int tid  = threadIdx.x;
  const int w    = tid >> 5;
  const int lane = tid & 31;
  const int ml   = lane & 15;
  const int lhi  = lane >> 4;                 // 0: K pair (0,1); 1: K pair (2,3)
  const int mwave = (w & 3) * 32;             // 4 waves along M
  const int nwave = (w >> 2) * 64;            // 2 waves along N

  __shared__ __align__(16) float As[KB / 2][TM][2];
  __shared__ __align__(16) float Bs[KB / 2][TN][2];
  __shared__ int   tokS[TM];
  __shared__ float xsumS[TM];
  __shared__ float ssS[TN];
  __shared__ float metS[TN];

  if (tid < TM) {
    int rr = r0 + tid;
    int tok = (rr < g1) ? perm[rr] : -1;
    tokS[tid]  = tok;
    xsumS[tid] = (tok >= 0) ? xsum[tok] : 0.f;
    ssS[tid]  = ssp[(size_t)n * OUT + jbase + tid];
    metS[tid] = metas[(size_t)n * OUT + jbase + tid];
  }
  __syncthreads();
  const int tok0 = tokS[0];                   // always valid (r0 < g1)

  v8f acc[2][4];
#pragma unroll
  for (int mb = 0; mb < 2; ++mb)
#pragma unroll
    for (int nb = 0; nb < 4; ++nb) acc[mb][nb] = (v8f)0.f;

  for (int kt = 0; kt < H; kt += KB) {
    // --- stage A: 128 gathered x rows x KB (pair-interleaved)
    {
      int row = tid >> 1;
      int kc  = (tid & 1) * 16;
      int tok = tokS[row];
      int tokL = (tok < 0) ? tok0 : tok;
      const float* xp = x + (size_t)tokL * H + kt + kc;
#pragma unroll
      for (int c = 0; c < 16; c += 4) {
        float4 v = *(const float4*)(xp + c);
        v2f lo, hi;
        lo.x = v.x; lo.y = v.y; hi.x = v.z; hi.y = v.w;
        *(v2f*)&As[(kc + c) >> 1][row][0]       = lo;
        *(v2f*)&As[((kc + c) >> 1) + 1][row][0] = hi;
      }
    }
    // --- stage B phase 1: base_weight^T (row-major bw: contiguous in k)
    {
      int j  = tid >> 1;
      int kc = (tid & 1) * 16;
      const float* bp = bw + (size_t)(jbase + j) * H + kt + kc;
#pragma unroll
      for (int c = 0; c < 16; c += 4) {
        float4 v = *(const float4*)(bp + c);
        v2f lo, hi;
        lo.x = v.x; lo.y = v.y; hi.x = v.z; hi.y = v.w;
        *(v2f*)&Bs[(kc + c) >> 1][j][0]       = lo;
        *(v2f*)&Bs[((kc + c) >> 1) + 1][j][0] = hi;
      }
    }
    __syncthreads();
    // --- stage B phase 2: += ss * DeltaW (contiguous in j, coalesced reads)
    {
      int k  = tid >> 3;
      int j0 = (tid & 7) * 16;
      const int kp = k >> 1, kb = k & 1;
      const float* dp = DW + ((size_t)n * H + kt + k) * OUT + jbase + j0;
#pragma unroll
      for (int c = 0; c < 16; c += 4) {
        float4 v = *(const float4*)(dp + c);
        Bs[kp][j0 + c + 0][kb] += ssS[j0 + c + 0] * v.x;
        Bs[kp][j0 + c + 1][kb] += ssS[j0 + c + 1] * v.y;
        Bs[kp][j0 + c + 2][kb] += ssS[j0 + c + 2] * v.z;
        Bs[kp][j0 + c + 3][kb] += ssS[j0 + c + 3] * v.w;
      }
      if (kt + KB < H) {   // global_prefetch_b8 of the next K-slab
        __builtin_prefetch(DW + ((size_t)n * H + kt + KB + k) * OUT + jbase + j0, 0, 0);
        __builtin_prefetch(bw + (size_t)(jbase + (tid >> 1)) * H + kt + KB + (tid & 1) * 16, 0, 0);
      }
    }
    __syncthreads();
    // --- compute: 8 k-steps x 8 fragments of v_wmma_f32_16x16x4_f32
#pragma unroll
    for (int kk = 0; kk < KB; kk += 4) {
      const int p = (kk >> 1) + lhi;          // K pair index for this lane half
      v2f af[2], bf[4];
#pragma unroll
      for (int mb = 0; mb < 2; ++mb)
        af[mb] = *(const v2f*)&As[p][mwave + mb * 16 + ml][0];
#pragma unroll
      for (int nb = 0; nb < 4; ++nb)
        bf[nb] = *(const v2f*)&Bs[p][nwave + nb * 16 + ml][0];
#pragma unroll
      for (int mb = 0; mb < 2; ++mb)
#pragma unroll
        for (int nb = 0; nb < 4; ++nb)
          acc[mb][nb] = wmma4(af[mb], bf[nb], acc[mb][nb]);
    }
    __syncthreads();
  }

  // --- LoRA tail: one extra K=16 slab (A = xa half, B = B_buffer[n])
  {
    const int half16 = (jbase >= BDIM) ? RANK : 0;
    if (tid < TM) {
      int tok = tokS[tid];
      int tokL = (tok < 0) ? tok0 : tok;
      const float* xp = xa + (size_t)tokL * 32 + half16;
#pragma unroll
      for (int k = 0; k < RANK; k += 2) {
        v2f t; t.x = xp[k]; t.y = xp[k + 1];
        *(v2f*)&As[k >> 1][tid][0] = t;
      }
    }
    {
      int k  = tid >> 4;           // 0..15
      int j0 = (tid & 15) * 8;     // 0..120
      const int kp = k >> 1, kb = k & 1;
      const float* bp = Bbuf + ((size_t)n * RANK + k) * OUT + jbase + j0;
#pragma unroll
      for (int c = 0; c < 8; c += 4) {
        float4 v = *(const float4*)(bp + c);
        Bs[kp][j0 + c + 0][kb] = v.x;
        Bs[kp][j0 + c + 1][kb] = v.y;
        Bs[kp][j0 + c + 2][kb] = v.z;
        Bs[kp][j0 + c + 3][kb] = v.w;
      }
    }
    __syncthreads();
#pragma unroll
    for (int kk = 0; kk < RANK; kk += 4) {
      const int p = (kk >> 1) + lhi;
      v2f af[2], bf[4];
#pragma unroll
      for (int mb = 0; mb < 2; ++mb)
        af[mb] = *(const v2f*)&As[p][mwave + mb * 16 + ml][0];
#pragma unroll
      for (int nb = 0; nb < 4; ++nb)
        bf[nb] = *(const v2f*)&Bs[p][nwave + nb * 16 + ml][0];
#pragma unroll
      for (int mb = 0; mb < 2; ++mb)
#pragma unroll
        for (int nb = 0; nb < 4; ++nb)
          acc[mb][nb] = wmma4(af[mb], bf[nb], acc[mb][nb]);
    }
  }

  // --- epilogue: out = acc - ss*metas*xsum, masked scatter by token
#pragma unroll
  for (int nb = 0; nb < 4; ++nb) {
    const int jj = nwave + nb * 16 + ml;
    const float csub = ssS[jj] * metS[jj];
    const size_t jg = (size_t)(jbase + jj);
#pragma unroll
    for (int mb = 0; mb < 2; ++mb) {
#pragma unroll
      for (int v = 0; v < 8; ++v) {
        int mloc = mwave + mb * 16 + v + 8 * lhi;
        int tok = tokS[mloc];
        if (tok >= 0)
          out[(size_t)tok * OUT + jg] = acc[mb][nb][v] - csub * xsumS[mloc];
      }
    }
  }
}

// ---------------------------------------------------------------- launcher
extern "C" void kernel_launch(void* const* d_in, const int* in_sizes, int n_in,
                              void* d_out, int out_size, void* d_ws, size_t ws_size,
                              hipStream_t stream) {
  const float* x     = (const float*)d_in[0];
  const int*   widx  = (const int*)d_in[1];
  const float* bw    = (const float*)d_in[2];
  const float* Abuf  = (const float*)d_in[3];
  const float* Bbuf  = (const float*)d_in[4];
  const float* DW    = (const float*)d_in[5];
  const float* metas = (const float*)d_in[6];
  const float* ssp   = (const float*)d_in[7];
  float* out = (float*)d_out;

  // workspace carve-up (~560 KB)
  int*   counts = (int*)d_ws;
  int*   offs   = counts + 8;
  int*   cursor = offs + 8;
  int*   perm   = cursor + 8;
  float* xsum   = (float*)(perm + S);
  float* xa     = xsum + S;

  k_zero<<<1, 64, 0, stream>>>(counts, cursor);
  k_count<<<S / 256, 256, 0, stream>>>(widx, counts);
  k_scan<<<1, 32, 0, stream>>>(counts, offs);
  k_scatter<<<S / 256, 256, 0, stream>>>(widx, offs, cursor, perm);
  k_prep<<<S, 256, 0, stream>>>(x, widx, Abuf, xa, xsum);

  dim3 grid(OUT / TN, S / TM, NAD);   // 64 x 32 x 4 tiles (empties exit early)
  k_main<<<grid, 256, 0, stream>>>(x, perm, offs, bw, Bbuf, DW, metas, ssp,
                                   xa, xsum, out);
}